// YOLOv7_62947040690682
// MI455X (gfx1250) — compile-verified
//
#include <hip/hip_runtime.h>
#include <stdint.h>

#define B_ 16
#define N_ 25200
#define NC_ 80
#define C_ 85                      // 5 + NC
#define CONF_THRES 0.25f
#define IOU_THRES 0.45f
#define MAX_CAND 1000
#define CLS_OFF 4096.0f
#define EPS_ 1e-7f

// ---------------------------------------------------------------------------
// Kernel A: streaming score pass (bandwidth-dominant: 137 MB read).
// Each block stages ABLK anchor rows (ABLK*85 floats) into LDS with
// global_load_async_to_lds_b128 (coalesced 16B/lane, ASYNCcnt-tracked),
// waits with s_wait_asynccnt, then each thread reduces the 80-class max
// for one anchor out of LDS (odd dword stride 85 -> LDS-bank conflict-free).
// ---------------------------------------------------------------------------
#define ABLK 128
#define TILE_FLOATS (ABLK * C_)        // 10880 floats
#define TILE_VEC4   (TILE_FLOATS / 4)  // 2720 x b128

__global__ __launch_bounds__(ABLK)
void score_kernel(const float* __restrict__ pred,
                  float* __restrict__ score,
                  int* __restrict__ clsidx)
{
    __shared__ float tile[TILE_FLOATS];          // 43,520 B
    const int tid = threadIdx.x;
    const long long tileBase = (long long)blockIdx.x * TILE_FLOATS;

    // async global -> LDS staging
    unsigned ldsBase = (unsigned)(uintptr_t)(&tile[0]); // low 32b = LDS offset
    const float* gbase = pred + tileBase;
    for (int k = tid; k < TILE_VEC4; k += ABLK) {
        unsigned lds_addr = ldsBase + (unsigned)k * 16u;
        unsigned long long gaddr = (unsigned long long)(uintptr_t)(gbase + k * 4);
        asm volatile("global_load_async_to_lds_b128 %0, %1, off"
                     :: "v"(lds_addr), "v"(gaddr) : "memory");
    }
    asm volatile("s_wait_asynccnt 0" ::: "memory");
    __syncthreads();

    // one anchor per thread
    const float* a = &tile[tid * C_];
    float obj = a[4];
    float m = a[5];
    int   mi = 0;
    #pragma unroll 4
    for (int c = 1; c < NC_; ++c) {
        float v = a[5 + c];
        if (v > m) { m = v; mi = c; }
    }
    float conf  = obj * m;
    bool  valid = (obj > CONF_THRES) && (conf > CONF_THRES);
    long long gi = (long long)blockIdx.x * ABLK + tid;
    score[gi]  = valid ? conf : -1.0f;
    clsidx[gi] = mi;
}

// ---------------------------------------------------------------------------
// Kernel B: per-batch top-1000, deterministic.
// 2048-bin histogram picks threshold bin T with count(bin > T) < 1000.
// ALL entries with bin >= T are collected into 2048 slots (a set -- atomic
// slot order is canonicalized by the sort, so replays are bit-identical),
// then a 2048-wide bitonic sort (key desc, index asc on ties) and the top
// 1000 are emitted.
// ---------------------------------------------------------------------------
#define NBINS 2048
#define NS    2048                    // sort width / collection capacity

__global__ __launch_bounds__(1024)
void topk_kernel(const float* __restrict__ score,
                 int* __restrict__ topIdx,
                 float* __restrict__ topScore)
{
    __shared__ int   hist[NBINS];     // 8 KB
    __shared__ float skey[NS];        // 8 KB
    __shared__ int   sidx[NS];        // 8 KB
    __shared__ int   sThresh, sCtr;

    const int b = blockIdx.x, tid = threadIdx.x;
    const float* s = score + (long long)b * N_;

    for (int i = tid; i < NBINS; i += 1024) hist[i] = 0;
    __syncthreads();

    for (int n = tid; n < N_; n += 1024) {
        float v = s[n];
        if (v > 0.0f) {
            int bin = (int)(v * (float)NBINS);
            bin = bin < 0 ? 0 : (bin >= NBINS ? NBINS - 1 : bin);
            atomicAdd(&hist[bin], 1);
        }
    }
    __syncthreads();

    if (tid == 0) {
        int acc = 0, t = -1;
        for (int bin = NBINS - 1; bin >= 0; --bin) {
            if (acc + hist[bin] >= MAX_CAND) { t = bin; break; }
            acc += hist[bin];
        }
        sThresh = t;                 // -1 => fewer than MAX_CAND valid
        sCtr    = 0;
    }
    __syncthreads();

    for (int i = tid; i < NS; i += 1024) { skey[i] = -1.0f; sidx[i] = 0x7fffffff; }
    __syncthreads();

    const int t = sThresh;
    for (int n = tid; n < N_; n += 1024) {
        float v = s[n];
        if (v > 0.0f) {
            int bin = (int)(v * (float)NBINS);
            bin = bin < 0 ? 0 : (bin >= NBINS ? NBINS - 1 : bin);
            if (bin >= t) {          // t==-1: collect everything valid
                int p = atomicAdd(&sCtr, 1);
                if (p < NS) { skey[p] = v; sidx[p] = n; }
            }
        }
    }
    __syncthreads();

    // bitonic sort, NS elems, 1024 threads x 2: desc by key, asc index ties
    for (int k = 2; k <= NS; k <<= 1) {
        for (int j = k >> 1; j > 0; j >>= 1) {
            for (int e = tid; e < NS; e += 1024) {
                int ixj = e ^ j;
                if (ixj > e) {
                    float a = skey[e], c = skey[ixj];
                    int   ia = sidx[e], ic = sidx[ixj];
                    bool aAfter = (a < c) || (a == c && ia > ic);
                    bool desc   = ((e & k) == 0);
                    if (desc ? aAfter : !aAfter) {
                        skey[e] = c; skey[ixj] = a;
                        sidx[e] = ic; sidx[ixj] = ia;
                    }
                }
            }
            __syncthreads();
        }
    }

    if (tid < MAX_CAND) {
        topIdx[b * MAX_CAND + tid]   = sidx[tid];
        topScore[b * MAX_CAND + tid] = skey[tid];
    }
}

// ---------------------------------------------------------------------------
// Kernel C: gather candidates, xywh -> xyxy, class-offset boxes.
// ---------------------------------------------------------------------------
__global__ __launch_bounds__(256)
void gather_kernel(const float* __restrict__ pred,
                   const int* __restrict__ clsidx,
                   const int* __restrict__ topIdx,
                   const float* __restrict__ topScore,
                   float4* __restrict__ candRaw,
                   float4* __restrict__ candOff,
                   float* __restrict__ candConf,
                   float* __restrict__ candCls)
{
    int tI = blockIdx.x * blockDim.x + threadIdx.x;
    if (tI >= B_ * MAX_CAND) return;
    int b = tI / MAX_CAND;

    float sc   = topScore[tI];
    int   idx  = topIdx[tI];
    bool  ok   = (sc > 0.0f) && (idx >= 0) && (idx < N_);
    int   gi   = ok ? idx : 0;

    const float* p = pred + ((long long)b * N_ + gi) * C_;
    float x = p[0], y = p[1], w = p[2], h = p[3];
    float x1 = x - w * 0.5f, y1 = y - h * 0.5f;
    float x2 = x + w * 0.5f, y2 = y + h * 0.5f;

    int   cls = clsidx[(long long)b * N_ + gi];
    float off = (float)cls * CLS_OFF;

    candRaw[tI] = make_float4(x1, y1, x2, y2);
    candOff[tI] = make_float4(x1 + off, y1 + off, x2 + off, y2 + off);
    candConf[tI] = sc;
    candCls[tI]  = (float)cls;
}

// ---------------------------------------------------------------------------
// Kernel D: sequential greedy NMS per batch (exact fori_loop semantics:
// barrier per i, keep[i] read after the barrier), then masked output write.
// 16 concurrent blocks overlap the 1000-step serial chains.
// ---------------------------------------------------------------------------
__global__ __launch_bounds__(1024)
void nms_kernel(const float4* __restrict__ candRaw,
                const float4* __restrict__ candOff,
                const float* __restrict__ candConf,
                const float* __restrict__ candCls,
                const float* __restrict__ topScore,
                float* __restrict__ out)
{
    __shared__ float bx[MAX_CAND * 4];       // 16,000 B
    __shared__ unsigned char keep[MAX_CAND]; //  1,000 B

    const int b = blockIdx.x, tid = threadIdx.x;
    const long long base = (long long)b * MAX_CAND;

    for (int i = tid; i < MAX_CAND; i += 1024) {
        float4 v = candOff[base + i];
        bx[i * 4 + 0] = v.x; bx[i * 4 + 1] = v.y;
        bx[i * 4 + 2] = v.z; bx[i * 4 + 3] = v.w;
        keep[i] = (topScore[base + i] > 0.0f) ? 1 : 0;
    }
    __syncthreads();

    float mx1 = 0.f, my1 = 0.f, mx2 = 0.f, my2 = 0.f, myArea = 0.f;
    if (tid < MAX_CAND) {
        mx1 = bx[tid * 4 + 0]; my1 = bx[tid * 4 + 1];
        mx2 = bx[tid * 4 + 2]; my2 = bx[tid * 4 + 3];
        myArea = (mx2 - mx1) * (my2 - my1);
    }

    for (int i = 0; i < MAX_CAND - 1; ++i) {
        if (tid > i && tid < MAX_CAND && keep[i] && keep[tid]) {
            float ax1 = bx[i * 4 + 0], ay1 = bx[i * 4 + 1];
            float ax2 = bx[i * 4 + 2], ay2 = bx[i * 4 + 3];
            float iw = fminf(ax2, mx2) - fmaxf(ax1, mx1);
            float ih = fminf(ay2, my2) - fmaxf(ay1, my1);
            iw = fmaxf(iw, 0.0f); ih = fmaxf(ih, 0.0f);
            float inter = iw * ih;
            float areaI = (ax2 - ax1) * (ay2 - ay1);
            float iou = inter / (areaI + myArea - inter + EPS_);
            if (iou > IOU_THRES) keep[tid] = 0;
        }
        __syncthreads();
    }

    if (tid < MAX_CAND) {
        long long o = base + tid;
        bool k = keep[tid] != 0;
        float4 rb = candRaw[o];
        out[o * 6 + 0] = k ? rb.x : 0.0f;
        out[o * 6 + 1] = k ? rb.y : 0.0f;
        out[o * 6 + 2] = k ? rb.z : 0.0f;
        out[o * 6 + 3] = k ? rb.w : 0.0f;
        out[o * 6 + 4] = k ? candConf[o] : 0.0f;
        out[o * 6 + 5] = k ? candCls[o]  : 0.0f;
        out[(long long)B_ * MAX_CAND * 6 + o] = k ? 1.0f : 0.0f;
    }
}

// ---------------------------------------------------------------------------
extern "C" void kernel_launch(void* const* d_in, const int* in_sizes, int n_in,
                              void* d_out, int out_size, void* d_ws, size_t ws_size,
                              hipStream_t stream)
{
    (void)in_sizes; (void)n_in; (void)out_size; (void)ws_size;
    const float* pred = (const float*)d_in[0];
    float* out = (float*)d_out;

    char* ws = (char*)d_ws;
    float*  score    = (float*) (ws + 0);          // 403200 f = 1,612,800 B
    int*    clsidx   = (int*)   (ws + 1612800);    // 403200 i = 1,612,800 B
    int*    topIdx   = (int*)   (ws + 3225600);    // 16000 i
    float*  topScore = (float*) (ws + 3289600);    // 16000 f
    float4* candRaw  = (float4*)(ws + 3353600);    // 16000 * float4
    float4* candOff  = (float4*)(ws + 3609600);    // 16000 * float4
    float*  candConf = (float*) (ws + 3865600);    // 16000 f
    float*  candCls  = (float*) (ws + 3929600);    // 16000 f

    const int nBlocksA = (B_ * N_) / ABLK;         // 403200/128 = 3150
    score_kernel<<<nBlocksA, ABLK, 0, stream>>>(pred, score, clsidx);

    topk_kernel<<<B_, 1024, 0, stream>>>(score, topIdx, topScore);

    const int nG = (B_ * MAX_CAND + 255) / 256;    // 63
    gather_kernel<<<nG, 256, 0, stream>>>(pred, clsidx, topIdx, topScore,
                                          candRaw, candOff, candConf, candCls);

    nms_kernel<<<B_, 1024, 0, stream>>>(candRaw, candOff, candConf, candCls,
                                        topScore, out);
}